// CliffordEngine_4157528342663
// MI455X (gfx1250) — compile-verified
//
#include <hip/hip_runtime.h>

// CDNA5 / gfx1250: wave32, WMMA f32 16x16x4, async global->LDS staging.
typedef __attribute__((ext_vector_type(2))) float v2f;
typedef __attribute__((ext_vector_type(4))) float v4f;
typedef __attribute__((ext_vector_type(8))) float v8f;

#define NCOMP 256
#define PAIRS 1024          // 4 * 256
#define WAVES_PER_BLOCK 8

// Reordering sign of e_a * e_b in Cl(4,0): (-1)^{sum over set bits p of a of popc(b & below(p))}
__device__ __forceinline__ float sigma4(int a, int b) {
  int cnt = 0;
#pragma unroll
  for (int p = 0; p < 4; ++p)
    if ((a >> p) & 1) cnt += __popc(b & ((1 << p) - 1));
  return (cnt & 1) ? -1.0f : 1.0f;
}

__global__ __launch_bounds__(256)
void clifford_wmma_kernel(const float* __restrict__ A,
                          const float* __restrict__ B,
                          float* __restrict__ out) {
  // sigma4/mask table: per (a,b) in 16x16: {gp, wedge(a&b==0), lc(a subset b), rc(b subset a)} * sign
  __shared__ v4f   sT[256];
  __shared__ float sA[WAVES_PER_BLOCK][NCOMP];  // raw A, then A-hat after sign pass
  __shared__ float sB[WAVES_PER_BLOCK][NCOMP];  // raw B rows

  const int tid  = threadIdx.x;
  const int lane = tid & 31;
  const int wave = tid >> 5;
  const int pair = blockIdx.x * WAVES_PER_BLOCK + wave;
  const float* Ap = A + (size_t)pair * NCOMP;
  const float* Bp = B + (size_t)pair * NCOMP;

  // --- Async global->LDS staging (ASYNCcnt path): 1KB A + 1KB B per wave -----
  {
    const unsigned voff = (unsigned)lane * 16u;                    // 32 lanes x 16B = 512B per op
    const unsigned va = (unsigned)(uintptr_t)&sA[wave][0] + voff;  // LDS byte address (low 32 bits)
    const unsigned vb = (unsigned)(uintptr_t)&sB[wave][0] + voff;
    // INST_OFFSET applies to both LDS and global sides (ISA 08 §4.4)
    asm volatile("global_load_async_to_lds_b128 %0, %1, %2 offset:0"   :: "v"(va), "v"(voff), "s"(Ap) : "memory");
    asm volatile("global_load_async_to_lds_b128 %0, %1, %2 offset:512" :: "v"(va), "v"(voff), "s"(Ap) : "memory");
    asm volatile("global_load_async_to_lds_b128 %0, %1, %2 offset:0"   :: "v"(vb), "v"(voff), "s"(Bp) : "memory");
    asm volatile("global_load_async_to_lds_b128 %0, %1, %2 offset:512" :: "v"(vb), "v"(voff), "s"(Bp) : "memory");
  }

  // --- Build sigma/mask table while the async copy is in flight --------------
  {
    int a = tid >> 4, b = tid & 15;
    float s  = sigma4(a, b);
    float we = ((a & b) == 0)        ? s : 0.0f;
    float lc = ((a & ~b & 15) == 0)  ? s : 0.0f;  // a subset of b
    float rc = ((b & ~a & 15) == 0)  ? s : 0.0f;  // b subset of a
    v4f t = {s, we, lc, rc};
    sT[tid] = t;
  }

  asm volatile("s_wait_asynccnt 0" ::: "memory");

  // --- Diagonal term + in-place A-hat sign pass (lane-private 8-elem slab) ---
  float diag = 0.0f;
  {
    float* aw = sA[wave];
    float* bw = sB[wave];
    const int base = lane * 8;
    v4f a0 = *(v4f*)&aw[base];     v4f a1 = *(v4f*)&aw[base + 4];
    v4f b0 = *(v4f*)&bw[base];     v4f b1 = *(v4f*)&bw[base + 4];
    v4f s0, s1;
#pragma unroll
    for (int e = 0; e < 8; ++e) {
      const int j  = base + e;
      const int jh = j >> 4, jl = j & 15;
      const float sa = ((__popc(jh) & __popc(jl)) & 1) ? -1.0f : 1.0f;  // (-1)^{popc(jh)popc(jl)}
      const int g = __popc(j);
      const float sj = (((g * (g - 1)) >> 1) & 1) ? -1.0f : 1.0f;       // sigma8(j,j)
      const float av = (e < 4) ? a0[e] : a1[e - 4];
      const float bv = (e < 4) ? b0[e] : b1[e - 4];
      diag += sj * av * bv;
      if (e < 4) s0[e] = av * sa; else s1[e - 4] = av * sa;
    }
    *(v4f*)&aw[base]     = s0;
    *(v4f*)&aw[base + 4] = s1;
  }
  __syncthreads();
#pragma unroll
  for (int off = 16; off > 0; off >>= 1)
    diag += __shfl_xor(diag, off, 32);

  const int m    = lane & 15;   // A operand: M row; B operand / D: N col
  const int half = lane >> 4;   // K pair group (k = 2*half + {0,1}) per ISA 16x4 layout
  const int pm   = __popc(m) & 1;
  const bool modd = (m & 1) != 0;  // parity select for merged B pair loads

  v8f acc_gp = {}, acc_we = {}, acc_lc = {}, acc_rc = {};

  for (int kh = 0; kh < 16; ++kh) {
    const int jh = m ^ kh;                        // source A row feeding out-row m
    const v4f h  = sT[jh * 16 + kh];              // high-nibble coefficient per family
    const float twist = ((__popc(kh) & 1) & pm) ? -1.0f : 1.0f;  // (-1)^{popc(kh)popc(il)}
    const float* Arow = &sA[wave][jh * 16];
    const float* Brow = &sB[wave][kh * 16];

#pragma unroll
    for (int s = 0; s < 4; ++s) {
      const int k0 = 4 * s + 2 * half;            // global K for VGPR0 of this slice
      const int k1 = k0 + 1;
      const float a0 = Arow[k0], a1 = Arow[k1];
      const int kl0 = m ^ k0;                     // kl = il ^ jl; kl1 = kl0 ^ 1 (adjacent pair)
      const v2f bp = *(const v2f*)&Brow[kl0 & ~1];
      const float b0 = (modd ? bp.y : bp.x) * twist;
      const float b1 = (modd ? bp.x : bp.y) * twist;
      const v4f l0 = sT[k0 * 16 + kl0];
      const v4f l1 = sT[k1 * 16 + (kl0 ^ 1)];

      v2f a, b;
      a.x = a0 * h.x; a.y = a1 * h.x; b.x = b0 * l0.x; b.y = b1 * l1.x;
      acc_gp = __builtin_amdgcn_wmma_f32_16x16x4_f32(false, a, false, b, (short)0, acc_gp, false, false);
      a.x = a0 * h.y; a.y = a1 * h.y; b.x = b0 * l0.y; b.y = b1 * l1.y;
      acc_we = __builtin_amdgcn_wmma_f32_16x16x4_f32(false, a, false, b, (short)0, acc_we, false, false);
      a.x = a0 * h.z; a.y = a1 * h.z; b.x = b0 * l0.z; b.y = b1 * l1.z;
      acc_lc = __builtin_amdgcn_wmma_f32_16x16x4_f32(false, a, false, b, (short)0, acc_lc, false, false);
      a.x = a0 * h.w; a.y = a1 * h.w; b.x = b0 * l0.w; b.y = b1 * l1.w;
      acc_rc = __builtin_amdgcn_wmma_f32_16x16x4_f32(false, a, false, b, (short)0, acc_rc, false, false);
    }
  }

  // --- Epilogue: output twist T(ih,il) = (-1)^{popc(ih)popc(il)}; inner = lc + rc - diag@0.
  // Streamed outputs -> non-temporal stores.
  float* out_gp = out;
  float* out_we = out + (size_t)PAIRS * NCOMP;
  float* out_in = out + 2 * (size_t)PAIRS * NCOMP;
  const size_t base = (size_t)pair * NCOMP;
#pragma unroll
  for (int v = 0; v < 8; ++v) {
    const int ih = v + 8 * half;
    const int i  = ih * 16 + m;
    const float T = ((__popc(ih) & 1) & pm) ? -1.0f : 1.0f;
    float g  = acc_gp[v] * T;
    float w  = acc_we[v] * T;
    float in = (acc_lc[v] + acc_rc[v]) * T;
    if (i == 0) in -= diag;   // remove double-counted scalar diagonal (after WMMAs)
    __builtin_nontemporal_store(g,  &out_gp[base + i]);
    __builtin_nontemporal_store(w,  &out_we[base + i]);
    __builtin_nontemporal_store(in, &out_in[base + i]);
  }
}

extern "C" void kernel_launch(void* const* d_in, const int* in_sizes, int n_in,
                              void* d_out, int out_size, void* d_ws, size_t ws_size,
                              hipStream_t stream) {
  (void)in_sizes; (void)n_in; (void)d_ws; (void)ws_size; (void)out_size;
  const float* A = (const float*)d_in[0];
  const float* B = (const float*)d_in[1];
  // index/sign arrays d_in[2..13] are unused: signs/masks are recomputed analytically on device
  float* out = (float*)d_out;
  clifford_wmma_kernel<<<PAIRS / WAVES_PER_BLOCK, 256, 0, stream>>>(A, B, out);
}